// ResidualGCNBlock_83915071030244
// MI455X (gfx1250) — compile-verified
//
#include <hip/hip_runtime.h>
#include <hip/hip_bf16.h>

// ---------------------------------------------------------------------------
// ResidualGCNBlock for MI455X (gfx1250, wave32)
//   N=50000 nodes, E=800000 edges, D_IN=64, D_OUT=128
// GEMMs via V_WMMA_F32_16X16X4_F32 (fp32 matrix pipe); edge scatter via
// L2-resident global_atomic_add_f32 (xw/agg are 25.6MB each -> live in 192MB L2)
// ---------------------------------------------------------------------------

#define DIN 64
#define DOUT 128

typedef float v2f __attribute__((ext_vector_type(2)));
typedef float v8f __attribute__((ext_vector_type(8)));

// --- K0: deg = 1.0 (self loop contribution) --------------------------------
__global__ __launch_bounds__(256) void k_init_deg(float* __restrict__ deg, int N) {
    int i = blockIdx.x * 256 + threadIdx.x;
    if (i < N) deg[i] = 1.0f;
}

// --- K1: degree accumulation over edge destinations ------------------------
__global__ __launch_bounds__(256) void k_deg_accum(const long long* __restrict__ dst,
                                                   float* __restrict__ deg, int E) {
    int e = blockIdx.x * 256 + threadIdx.x;
    if (e < E) {
        int d = (int)dst[e];
        unsafeAtomicAdd(&deg[d], 1.0f);
    }
}

// --- K2: dinv = rsqrt(deg)  (deg >= 1 always due to self loops) ------------
__global__ __launch_bounds__(256) void k_dinv(float* __restrict__ deg, int N) {
    int i = blockIdx.x * 256 + threadIdx.x;
    if (i < N) deg[i] = __frsqrt_rn(deg[i]);
}

// --- K3: WMMA GEMM: xw = x@W ; agg = xw*dinv^2 + b ; resid = x@res_W + res_b
// One 16x16 output tile per wave32; 8 waves (256 threads) per block.
// Tile id = row_tile*16 + col_tile, col_tile 0..7 -> W path, 8..15 -> res_W.
__global__ __launch_bounds__(256) void k_gemm(const float* __restrict__ x,
                                              const float* __restrict__ W,
                                              const float* __restrict__ resW,
                                              const float* __restrict__ b,
                                              const float* __restrict__ res_b,
                                              const float* __restrict__ dinv,
                                              float* __restrict__ xw,
                                              float* __restrict__ agg,
                                              float* __restrict__ resid,
                                              int N) {
    const int wave = threadIdx.x >> 5;
    const int lane = threadIdx.x & 31;
    const int n_row_tiles = (N + 15) >> 4;
    const int tile = blockIdx.x * 8 + wave;
    if (tile >= n_row_tiles * 16) return;

    const int row_tile = tile >> 4;
    const int ct       = tile & 15;
    const bool is_res  = (ct >= 8);
    const int m0 = row_tile << 4;
    const int n0 = (ct & 7) << 4;
    const float* __restrict__ Wm = is_res ? resW : W;

    const int half = lane >> 4;   // 0: K pair {0,1}, 1: K pair {2,3} within k-block
    const int mr   = lane & 15;   // row within A tile / column within B,C,D tile

    int arow = m0 + mr;
    if (arow >= N) arow = N - 1;  // clamp (N=50000 is 16-aligned; safety only)
    const float* __restrict__ xr = x + (size_t)arow * DIN;

    v8f c = {};
    #pragma unroll
    for (int kb = 0; kb < DIN; kb += 4) {
        // A 16x4 fp32 layout: v0 = K=(0|2), v1 = K=(1|3) selected by lane half
        v2f a, bb;
        a.x  = xr[kb + 2 * half + 0];
        a.y  = xr[kb + 2 * half + 1];
        // B 4x16 fp32 layout mirrors A: row K striped across lanes
        bb.x = Wm[(size_t)(kb + 2 * half + 0) * DOUT + n0 + mr];
        bb.y = Wm[(size_t)(kb + 2 * half + 1) * DOUT + n0 + mr];
        c = __builtin_amdgcn_wmma_f32_16x16x4_f32(
                /*neg_a=*/false, a, /*neg_b=*/false, bb,
                /*c_mod=*/(short)0, c, /*reuse_a=*/false, /*reuse_b=*/false);
    }

    const int col = n0 + mr;
    if (!is_res) {
        const float bc = b[col];
        #pragma unroll
        for (int v = 0; v < 8; ++v) {
            const int row = m0 + v + 8 * half;   // C/D layout: vgpr v -> M=v (+8 for hi half)
            if (row < N) {
                const float val = c[v];
                xw[(size_t)row * DOUT + col] = val;
                const float di = dinv[row];
                agg[(size_t)row * DOUT + col] = val * di * di + bc;  // self-loop + bias
            }
        }
    } else {
        const float rbc = res_b[col];
        #pragma unroll
        for (int v = 0; v < 8; ++v) {
            const int row = m0 + v + 8 * half;
            if (row < N) resid[(size_t)row * DOUT + col] = c[v] + rbc;
        }
    }
}

// --- K4: edge scatter: agg[dst] += xw[src] * (dinv[src]*dinv[dst]) ----------
// 32 lanes per edge, each lane: 1x float4 load + 4x global_atomic_add_f32.
// xw and agg are L2-resident (25.6MB each << 192MB L2).
__global__ __launch_bounds__(256) void k_scatter(const long long* __restrict__ src,
                                                 const long long* __restrict__ dst,
                                                 const float* __restrict__ dinv,
                                                 const float* __restrict__ xw,
                                                 float* __restrict__ agg, int E) {
    const long long tid = (long long)blockIdx.x * 256 + threadIdx.x;
    const int e    = (int)(tid >> 5);
    const int lane = (int)(tid & 31);
    if (e >= E) return;
    const int s = (int)src[e];
    const int d = (int)dst[e];
    const float nrm = dinv[s] * dinv[d];

    const float4 m = ((const float4*)(xw + (size_t)s * DOUT))[lane];
    float* __restrict__ ar = agg + (size_t)d * DOUT + lane * 4;
    unsafeAtomicAdd(ar + 0, m.x * nrm);
    unsafeAtomicAdd(ar + 1, m.y * nrm);
    unsafeAtomicAdd(ar + 2, m.z * nrm);
    unsafeAtomicAdd(ar + 3, m.w * nrm);
}

// --- K5: LayerNorm(128) + residual + SiLU; one wave32 per node --------------
__global__ __launch_bounds__(256) void k_finalize(const float* __restrict__ agg,
                                                  const float* __restrict__ resid,
                                                  const float* __restrict__ ln_w,
                                                  const float* __restrict__ ln_b,
                                                  float* __restrict__ out, int N) {
    const int wave = threadIdx.x >> 5;
    const int lane = threadIdx.x & 31;
    const int node = blockIdx.x * 8 + wave;
    if (node >= N) return;

    const float4 a = ((const float4*)(agg + (size_t)node * DOUT))[lane];

    float s = a.x + a.y + a.z + a.w;
    #pragma unroll
    for (int off = 16; off > 0; off >>= 1) s += __shfl_xor(s, off, 32);
    const float mu = s * (1.0f / (float)DOUT);

    const float d0 = a.x - mu, d1 = a.y - mu, d2 = a.z - mu, d3 = a.w - mu;
    float v = d0 * d0 + d1 * d1 + d2 * d2 + d3 * d3;
    #pragma unroll
    for (int off = 16; off > 0; off >>= 1) v += __shfl_xor(v, off, 32);
    const float rstd = __frsqrt_rn(v * (1.0f / (float)DOUT) + 1e-5f);

    const int c0 = lane * 4;
    const float4 lw = *(const float4*)(ln_w + c0);
    const float4 lb = *(const float4*)(ln_b + c0);
    const float4 r  = ((const float4*)(resid + (size_t)node * DOUT))[lane];

    float4 y;
    y.x = d0 * rstd * lw.x + lb.x + r.x;
    y.y = d1 * rstd * lw.y + lb.y + r.y;
    y.z = d2 * rstd * lw.z + lb.z + r.z;
    y.w = d3 * rstd * lw.w + lb.w + r.w;

    float4 o;
    o.x = y.x / (1.0f + __expf(-y.x));
    o.y = y.y / (1.0f + __expf(-y.y));
    o.z = y.z / (1.0f + __expf(-y.z));
    o.w = y.w / (1.0f + __expf(-y.w));

    ((float4*)(out + (size_t)node * DOUT))[lane] = o;
}

// ---------------------------------------------------------------------------
extern "C" void kernel_launch(void* const* d_in, const int* in_sizes, int n_in,
                              void* d_out, int out_size, void* d_ws, size_t ws_size,
                              hipStream_t stream) {
    const float*     x     = (const float*)d_in[0];
    const long long* ei    = (const long long*)d_in[1];   // int64 [2, E]
    const float*     W     = (const float*)d_in[2];
    const float*     b     = (const float*)d_in[3];
    const float*     ln_w  = (const float*)d_in[4];
    const float*     ln_b  = (const float*)d_in[5];
    const float*     resW  = (const float*)d_in[6];
    const float*     res_b = (const float*)d_in[7];
    float*           out   = (float*)d_out;

    const int N = in_sizes[0] / DIN;
    const int E = in_sizes[1] / 2;

    const long long* src = ei;       // edge_index[0]
    const long long* dst = ei + E;   // edge_index[1]

    // Workspace layout: dinv[N] | xw[N*128] | agg[N*128] | resid[N*128]
    char*  ws   = (char*)d_ws;
    float* dinv = (float*)ws;
    size_t off  = (((size_t)N * 4) + 255) & ~(size_t)255;
    float* xw    = (float*)(ws + off); off += (size_t)N * DOUT * 4;
    float* agg   = (float*)(ws + off); off += (size_t)N * DOUT * 4;
    float* resid = (float*)(ws + off);

    k_init_deg<<<(N + 255) / 256, 256, 0, stream>>>(dinv, N);
    k_deg_accum<<<(E + 255) / 256, 256, 0, stream>>>(dst, dinv, E);
    k_dinv<<<(N + 255) / 256, 256, 0, stream>>>(dinv, N);

    const int n_tiles = ((N + 15) / 16) * 16;   // row tiles x 16 column tiles
    k_gemm<<<(n_tiles + 7) / 8, 256, 0, stream>>>(x, W, resW, b, res_b, dinv,
                                                  xw, agg, resid, N);

    const long long scat_threads = (long long)E * 32;
    k_scatter<<<(int)((scat_threads + 255) / 256), 256, 0, stream>>>(src, dst, dinv,
                                                                     xw, agg, E);

    k_finalize<<<(N + 7) / 8, 256, 0, stream>>>(agg, resid, ln_w, ln_b, out, N);
}